// WordnnEmbedding_21345987461491
// MI455X (gfx1250) — compile-verified
//
#include <hip/hip_runtime.h>

#define B_ 4
#define N_ 128
#define H_ 512
#define W_ 512
#define VOCAB_ 30522
#define HIDDEN_ 768
#define EDIM_ 64

typedef float v2f __attribute__((ext_vector_type(2)));
typedef float v8f __attribute__((ext_vector_type(8)));

// ---------------------------------------------------------------------------
// Kernel 1: table[v][e] = sum_h emb[v][h] * proj[e][h]
// M = VOCAB, N = EDIM(64), K = HIDDEN(768), via V_WMMA_F32_16X16X4_F32.
// Block = 128 threads = 4 waves. Wave `w` computes row-tile mt = bx*4+w:
// rows [16*mt, 16*mt+16) x all 64 cols (4 accumulators of 8 VGPRs).
// A 16x4 layout: lanes 0-15 -> K = k0,k0+1 ; lanes 16-31 -> K = k0+2,k0+3.
// B 4x16 layout: VGPR0 = K rows {k0, k0+2}, VGPR1 = {k0+1, k0+3}, N = lane&15.
// ---------------------------------------------------------------------------
__global__ __launch_bounds__(128) void table_gemm_wmma(
    const float* __restrict__ emb, const float* __restrict__ proj,
    float* __restrict__ table) {
  const int lane = threadIdx.x & 31;
  const int wave = threadIdx.x >> 5;
  const int mt = blockIdx.x * 4 + wave;
  const int rlane = lane & 15;
  const int kh = lane >> 4;          // 0: lanes 0-15, 1: lanes 16-31

  int row = mt * 16 + rlane;
  int rc = row < VOCAB_ ? row : (VOCAB_ - 1);   // clamp: keep EXEC all-1s

  const float* __restrict__ arow = emb + (size_t)rc * HIDDEN_;
  const float* __restrict__ bp0 = proj + (size_t)(rlane + 0) * HIDDEN_;
  const float* __restrict__ bp1 = proj + (size_t)(rlane + 16) * HIDDEN_;
  const float* __restrict__ bp2 = proj + (size_t)(rlane + 32) * HIDDEN_;
  const float* __restrict__ bp3 = proj + (size_t)(rlane + 48) * HIDDEN_;

  v8f c0 = {}; v8f c1 = {}; v8f c2 = {}; v8f c3 = {};

  for (int k0 = 0; k0 < HIDDEN_; k0 += 4) {
    const int k = k0 + 2 * kh;
    v2f a  = *(const v2f*)(arow + k);
    v2f b0 = *(const v2f*)(bp0 + k);
    v2f b1 = *(const v2f*)(bp1 + k);
    v2f b2 = *(const v2f*)(bp2 + k);
    v2f b3 = *(const v2f*)(bp3 + k);
    c0 = __builtin_amdgcn_wmma_f32_16x16x4_f32(false, a, false, b0, (short)0, c0, false, false);
    c1 = __builtin_amdgcn_wmma_f32_16x16x4_f32(false, a, false, b1, (short)0, c1, false, false);
    c2 = __builtin_amdgcn_wmma_f32_16x16x4_f32(false, a, false, b2, (short)0, c2, false, false);
    c3 = __builtin_amdgcn_wmma_f32_16x16x4_f32(false, a, false, b3, (short)0, c3, false, false);
  }

  // C/D 16x16 layout: VGPR j holds M = j (lanes 0-15) / j+8 (lanes 16-31), N = lane&15.
#pragma unroll
  for (int j = 0; j < 8; ++j) {
    int r = mt * 16 + j + 8 * kh;
    if (r < VOCAB_) {
      float* trow = table + (size_t)r * EDIM_ + rlane;
      trow[0]  = c0[j];
      trow[16] = c1[j];
      trow[32] = c2[j];
      trow[48] = c3[j];
    }
  }
}

// ---------------------------------------------------------------------------
// Kernel 2: fused winner-rasterize + embedding gather.
// One thread per pixel; bbox (rounded to int) + token ids staged in LDS.
// Scan words from n=N-1 down: first covering word has max priority.
// winner==none -> id 0 -> table row 0 (matches ids_ext zero-prepend).
// Writes out[b][e][h][w]: 64 channel-strided stores, wave-contiguous in w.
// ---------------------------------------------------------------------------
__global__ __launch_bounds__(128) void winner_gather(
    const float* __restrict__ bbox, const int* __restrict__ ids,
    const float* __restrict__ table, float* __restrict__ out) {
  __shared__ int4 sbox[N_];
  __shared__ int sid[N_];

  const int b = blockIdx.z;
  const int h = blockIdx.y;
  const int w = blockIdx.x * 128 + threadIdx.x;

  if (threadIdx.x < N_) {
    const float* bb = bbox + ((size_t)b * N_ + threadIdx.x) * 4;
    int4 v;
    v.x = (int)__builtin_rintf(bb[0]);   // w0 (round-half-even, matches jnp.round)
    v.y = (int)__builtin_rintf(bb[1]);   // h0
    v.z = (int)__builtin_rintf(bb[2]);   // w1
    v.w = (int)__builtin_rintf(bb[3]);   // h1
    sbox[threadIdx.x] = v;
    sid[threadIdx.x] = ids[(size_t)b * N_ + threadIdx.x];
  }
  __syncthreads();

  int id = 0;
  for (int n = N_ - 1; n >= 0; --n) {
    int4 v = sbox[n];
    if (w >= v.x && w < v.z && h >= v.y && h < v.w) { id = sid[n]; break; }
  }

  const float* __restrict__ row = table + (size_t)id * EDIM_;
  const size_t hw = (size_t)H_ * W_;
  float* op = out + (size_t)b * EDIM_ * hw + (size_t)h * W_ + w;

#pragma unroll
  for (int e = 0; e < EDIM_; e += 4) {
    float4 v = *(const float4*)(row + e);   // 16B-aligned, L2-resident table
    op[(size_t)(e + 0) * hw] = v.x;
    op[(size_t)(e + 1) * hw] = v.y;
    op[(size_t)(e + 2) * hw] = v.z;
    op[(size_t)(e + 3) * hw] = v.w;
  }
}

// ---------------------------------------------------------------------------
extern "C" void kernel_launch(void* const* d_in, const int* in_sizes, int n_in,
                              void* d_out, int out_size, void* d_ws, size_t ws_size,
                              hipStream_t stream) {
  (void)in_sizes; (void)n_in; (void)out_size; (void)ws_size;
  // setup_inputs order: img, bbox, emb_weight, proj_weight, input_ids, stride
  const float* bbox = (const float*)d_in[1];
  const float* emb  = (const float*)d_in[2];
  const float* proj = (const float*)d_in[3];
  const int*   ids  = (const int*)d_in[4];
  // stride == 1 in this problem instance (d_in[5] ignored).

  float* table = (float*)d_ws;           // VOCAB_*EDIM_ floats = ~7.8 MB scratch
  float* out   = (float*)d_out;          // (B, 64, H, W) fp32

  const int mtiles  = (VOCAB_ + 15) / 16;   // 1908
  const int gblocks = (mtiles + 3) / 4;     // 477 blocks * 4 waves

  table_gemm_wmma<<<dim3(gblocks), dim3(128), 0, stream>>>(emb, proj, table);
  winner_gather<<<dim3(W_ / 128, H_, B_), dim3(128), 0, stream>>>(bbox, ids, table, out);
}